// photoreceptor_REIKE_18442589569721
// MI455X (gfx1250) — compile-verified
//
#include <hip/hip_runtime.h>

// ---------------------------------------------------------------------------
// Rieke photoreceptor ODE scan for MI455X (gfx1250, wave32).
// One thread per (batch, photoreceptor) chain; serial loop over T timesteps.
//   * 32-deep register load pipeline (split s_wait_loadcnt hides L2/HBM latency
//     in the serial chain; only ~1-2 waves/SIMD are available for this problem)
//   * global_prefetch_b8 at 96-step lookahead into the 192MB L2
//   * non-temporal stores for the 160MB write stream
//   * v_rcp/v_log/v_exp hardware transcendentals + integer-exponent fast paths
//   * 64-thread blocks: 625 workgroups spread the 1250-wave budget across WGPs
// No WMMA: the operator is a nonlinear time recurrence with zero matrix
// structure; tensor ops cannot be used without fabricating work.
// ---------------------------------------------------------------------------

#define DEVINL __device__ __forceinline__

DEVINL float fexp2(float v) { return __builtin_amdgcn_exp2f(v); }
DEVINL float flog2(float v) { return __builtin_amdgcn_logf(v); }   // log2
DEVINL float frcp (float v) { return __builtin_amdgcn_rcpf(v); }
DEVINL float fpow (float v, float e) { return fexp2(e * flog2(v)); }

// Problem geometry fixed by the reference: x is (B=16, T=1000, P=2500) f32.
constexpr int T      = 1000;
constexpr int P      = 2500;
constexpr int STEPS  = T - 1;                 // 999 scan steps
constexpr int PF     = 32;                    // register pipeline depth
constexpr int NCHUNK = STEPS / PF;            // 31 full chunks (992 steps)
constexpr int REM    = STEPS - NCHUNK * PF;   // 7 tail steps
constexpr int PDIST  = 96;                    // prefetch distance (timesteps)

template <bool H3, bool H4>
DEVINL void rieke_chain(const float* __restrict__ xp, float* __restrict__ op,
                        float sigma, float phi, float eta, float cgmp2cur,
                        float cgmphill, float cdark, float beta, float betaSlow,
                        float hillcoef, float hillaffinity, float gamma_eff,
                        float gdark)
{
    const float DT = 0.008f;  // 0.001 * timeBin(8)

    // Derived constants (match reference formulas; accurate powf, once/thread).
    const float darkCurrent = powf(gdark, cgmphill) * cgmp2cur * 0.5f;
    const float gdark2 = powf(2.0f * darkCurrent / cgmp2cur, 1.0f / cgmphill);
    const float cur2ca = beta * cdark / darkCurrent;
    const float smax = eta / phi * gdark2 *
                       (1.0f + powf(cdark / hillaffinity, hillcoef));

    // Initial state.
    float g  = gdark2;
    float s  = gdark2 * eta / phi;
    float c  = cdark;
    float cs = cdark;
    float r  = xp[0] * gamma_eff / sigma;
    float p  = (eta + r) / phi;

    // Per-step coefficients (hoisted).
    const float a_r  = 1.0f - DT * sigma;
    const float a_p  = 1.0f - DT * phi;
    const float dte  = DT * eta;
    const float a_c  = 1.0f - DT * beta;
    const float Kc   = DT * cur2ca * cgmp2cur;
    const float icd  = 1.0f / cdark;
    const float a_cs = 1.0f - DT * betaSlow;
    const float b_cs = DT * betaSlow;
    const float iha  = 1.0f / hillaffinity;
    const float osc  = -0.5f * cgmp2cur;

    auto hillG = [&](float v) -> float {
        if (H3) return v * v * v;                    // cgmphill == 3
        return fpow(v, cgmphill);
    };
    auto hillC = [&](float v) -> float {
        if (H4) { float w = v * v; return w * w; }   // hillcoef == 4
        return fpow(v, hillcoef);
    };

    float gpow = hillG(g);  // carry g^cgmphill across iterations (reused twice)

    // out[t=0] corresponds to g_all[0] == 0.
    __builtin_nontemporal_store(0.0f, op);
    float* oq = op + P;

    // Prime the register load pipeline with steps 0..PF-1.
    float xbuf[PF];
#pragma unroll
    for (int j = 0; j < PF; ++j) xbuf[j] = xp[(size_t)j * P];

    const float* xq = xp + (size_t)PF * P;      // next refill address
    const float* pq = xp + (size_t)PDIST * P;   // prefetch address

    auto body = [&](float xk) {
        float r1  = fmaf(a_r, r, gamma_eff * xk);                 // old r
        float p1  = fmaf(a_p, p, fmaf(DT, r, dte));               // old r
        float c1  = fmaf(a_c, c, Kc * gpow * frcp(fmaf(cs, icd, 1.0f)));
        float cs1 = fmaf(a_cs, cs, b_cs * c);                     // old c
        float s1  = smax * frcp(1.0f + hillC(c1 * iha));          // new c1
        float g1  = fmaf(DT, fmaf(-p, g, s), g);                  // old s, p
        float g1p = hillG(g1);
        __builtin_nontemporal_store(osc * g1p, oq);
        oq += P;
        r = r1; p = p1; c = c1; cs = cs1; s = s1; g = g1; gpow = g1p;
    };

    // Steady state: consume PF, refill PF (all refills in-bounds), prefetch.
    for (int ch = 0; ch < NCHUNK - 1; ++ch) {
#pragma unroll
        for (int i = 0; i < PF; ++i) {
            float xk = xbuf[i];
            xbuf[i] = xq[(size_t)i * P];
            __builtin_prefetch(pq, 0, 3);  // global_prefetch_b8
            pq += P;
            body(xk);
        }
        xq += (size_t)PF * P;
    }

    // Last full chunk: refill only the REM values the tail will consume.
#pragma unroll
    for (int i = 0; i < PF; ++i) {
        float xk = xbuf[i];
        if (i < REM) xbuf[i] = xq[(size_t)i * P];
        body(xk);
    }

    // Tail: drain remaining REM steps from registers (no loads).
#pragma unroll
    for (int i = 0; i < REM; ++i) body(xbuf[i]);
}

__global__ __launch_bounds__(64)
void rieke_pr_scan(const float* __restrict__ x, float* __restrict__ out, int N,
                   const float* __restrict__ g_sigma,
                   const float* __restrict__ g_phi,
                   const float* __restrict__ g_eta,
                   const float* __restrict__ g_cgmp2cur,
                   const float* __restrict__ g_cgmphill,
                   const float* __restrict__ g_cdark,
                   const float* __restrict__ g_beta,
                   const float* __restrict__ g_betaSlow,
                   const float* __restrict__ g_hillcoef,
                   const float* __restrict__ g_hillaffinity,
                   const float* __restrict__ g_gamma,
                   const float* __restrict__ g_gdark)
{
    int tid = blockIdx.x * blockDim.x + threadIdx.x;
    if (tid >= N) return;

    // Uniform parameter reads (single-element device arrays -> s_load).
    const float sigma        = g_sigma[0];
    const float phi          = g_phi[0];
    const float eta          = g_eta[0];
    const float cgmp2cur     = g_cgmp2cur[0];
    const float cgmphill     = g_cgmphill[0];
    const float cdark        = g_cdark[0];
    const float beta         = g_beta[0];
    const float betaSlow     = g_betaSlow[0];
    const float hillcoef     = g_hillcoef[0];
    const float hillaffinity = g_hillaffinity[0];
    const float gamma_eff    = g_gamma[0] * 0.125f;  // gamma / timeBin(8)
    const float gdark        = g_gdark[0];

    const int b  = tid / P;
    const int pp = tid - b * P;
    const float* xp = x   + (size_t)b * T * P + pp;
    float*       op = out + (size_t)b * T * P + pp;

    // Uniform dispatch: exponents live in device memory, so select the
    // integer-exponent fast paths with a (wave-uniform) runtime branch.
    const bool h3 = (cgmphill == 3.0f);
    const bool h4 = (hillcoef == 4.0f);

    if (h3 && h4)
        rieke_chain<true, true >(xp, op, sigma, phi, eta, cgmp2cur, cgmphill,
                                 cdark, beta, betaSlow, hillcoef, hillaffinity,
                                 gamma_eff, gdark);
    else if (h3)
        rieke_chain<true, false>(xp, op, sigma, phi, eta, cgmp2cur, cgmphill,
                                 cdark, beta, betaSlow, hillcoef, hillaffinity,
                                 gamma_eff, gdark);
    else if (h4)
        rieke_chain<false, true>(xp, op, sigma, phi, eta, cgmp2cur, cgmphill,
                                 cdark, beta, betaSlow, hillcoef, hillaffinity,
                                 gamma_eff, gdark);
    else
        rieke_chain<false, false>(xp, op, sigma, phi, eta, cgmp2cur, cgmphill,
                                  cdark, beta, betaSlow, hillcoef, hillaffinity,
                                  gamma_eff, gdark);
}

extern "C" void kernel_launch(void* const* d_in, const int* in_sizes, int n_in,
                              void* d_out, int out_size, void* d_ws, size_t ws_size,
                              hipStream_t stream)
{
    (void)n_in; (void)out_size; (void)d_ws; (void)ws_size;

    const float* x = (const float*)d_in[0];
    const int B = in_sizes[0] / (T * P);   // = 16 for the reference shapes
    const int N = B * P;                   // 40,000 independent chains

    dim3 block(64);
    dim3 grid((N + 63) / 64);
    rieke_pr_scan<<<grid, block, 0, stream>>>(
        x, (float*)d_out, N,
        (const float*)d_in[1],  (const float*)d_in[2],  (const float*)d_in[3],
        (const float*)d_in[4],  (const float*)d_in[5],  (const float*)d_in[6],
        (const float*)d_in[7],  (const float*)d_in[8],  (const float*)d_in[9],
        (const float*)d_in[10], (const float*)d_in[11], (const float*)d_in[12]);
}